// MMDLoss_26225070309743
// MI455X (gfx1250) — compile-verified
//
#include <hip/hip_runtime.h>
#include <hip/hip_bf16.h>

typedef __attribute__((ext_vector_type(2))) float v2f;
typedef __attribute__((ext_vector_type(8))) float v8f;

#define NROWS 4096
#define DIM   256
#define TILE_R 64
#define TILE_C 32
#define NTR   (NROWS / TILE_R)    // 64 row tiles
#define NTC   (NROWS / TILE_C)    // 128 col tiles
#define NTILES (NTR * NTC)        // 8192 waves per pair
#define WAVES_PER_BLOCK 8
#define BLOCKS_PER_PAIR (NTILES / WAVES_PER_BLOCK) // 1024
#define EPSK  1e-5f

// ---- workspace layout (floats) ----
// [0..2]   : d2 totals per pair (xx, yy, xy)
// [3..5]   : kernel-sum totals per pair
// [64..]   : sx (4096 row sq-norms of x)
// [4160..] : sy (4096 row sq-norms of y)
// [8256..] : block partials, 3 * 1024 floats
#define OFF_SX   64
#define OFF_SY   (64 + NROWS)
#define OFF_PART (OFF_SY + NROWS)

__device__ __forceinline__ float wave_reduce(float v) {
#pragma unroll
    for (int off = 16; off > 0; off >>= 1)
        v += __shfl_xor(v, off, 32);
    return v;
}

// Block reduce: 8 waves -> one float written by thread 0 to dst (deterministic).
__device__ __forceinline__ void block_reduce_store(float v, float* dst) {
    __shared__ float red[WAVES_PER_BLOCK];
    v = wave_reduce(v);
    const int lane = threadIdx.x & 31;
    const int wid  = threadIdx.x >> 5;
    if (lane == 0) red[wid] = v;
    __syncthreads();
    if (threadIdx.x == 0) {
        float t = 0.f;
#pragma unroll
        for (int i = 0; i < WAVES_PER_BLOCK; ++i) t += red[i];
        *dst = t;
    }
}

// One wave computes a 64x32 tile of G = A * B^T using V_WMMA_F32_16X16X4_F32.
// A fragment (16x4 f32): lane<16 -> {A[r+lane][k], A[r+lane][k+1]},
// lane>=16 -> {A[r+lane-16][k+2], A[r+lane-16][k+3]}. B (4x16) mirrors it, so
// both fragments load identically from row-major data.
// 8 accumulators (64 VGPRs) -> fits the 128-VGPR budget without spilling;
// 6 loads : 8 WMMAs per k-step, 8 independent accumulation chains.
__device__ __forceinline__ void gemm_tile(const float* __restrict__ A,
                                          const float* __restrict__ B,
                                          int r, int c, int lane,
                                          v8f acc[4][2]) {
    const int m    = lane & 15;
    const int half = lane >> 4;
    const float* ap = A + (size_t)(r + m) * DIM + 2 * half;
    const float* bp = B + (size_t)(c + m) * DIM + 2 * half;
    v8f z = {};
#pragma unroll
    for (int qa = 0; qa < 4; ++qa)
#pragma unroll
        for (int qb = 0; qb < 2; ++qb) acc[qa][qb] = z;

#pragma unroll 2
    for (int k = 0; k < DIM; k += 4) {
        v2f af[4], bf[2];
#pragma unroll
        for (int q = 0; q < 4; ++q)   // row-block stride 16*DIM = 16 KB -> imm offset
            af[q] = *(const v2f*)(ap + k + q * 16 * DIM);
#pragma unroll
        for (int q = 0; q < 2; ++q)
            bf[q] = *(const v2f*)(bp + k + q * 16 * DIM);
#pragma unroll
        for (int qa = 0; qa < 4; ++qa)
#pragma unroll
            for (int qb = 0; qb < 2; ++qb)
                acc[qa][qb] = __builtin_amdgcn_wmma_f32_16x16x4_f32(
                    false, af[qa], false, bf[qb], (short)0, acc[qa][qb],
                    false, false);
    }
}

__device__ __forceinline__ void select_pair(int z,
                                            const float* x, const float* y,
                                            const float* sx, const float* sy,
                                            const float*& A, const float*& B,
                                            const float*& sa, const float*& sb) {
    // z=0: (x,x)  z=1: (y,y)  z=2: (x,y)
    A  = (z == 1) ? y  : x;
    B  = (z == 0) ? x  : y;
    sa = (z == 1) ? sy : sx;
    sb = (z == 0) ? sx : sy;
}

// ---- row squared norms ----
__global__ void __launch_bounds__(256)
mmd_rownorm_kernel(const float* __restrict__ x,
                   const float* __restrict__ y,
                   float* __restrict__ ws) {
    const float* src = blockIdx.y ? y : x;
    float* dst = ws + (blockIdx.y ? OFF_SY : OFF_SX);
    const int row  = blockIdx.x * WAVES_PER_BLOCK + (threadIdx.x >> 5);
    const int lane = threadIdx.x & 31;
    const float* p = src + (size_t)row * DIM;
    float s = 0.f;
#pragma unroll
    for (int k = lane; k < DIM; k += 32) { float v = p[k]; s = fmaf(v, v, s); }
    s = wave_reduce(s);
    if (lane == 0) dst[row] = s;
}

// ---- pass 1: sum of clamped squared distances ----
__global__ void __launch_bounds__(256)
mmd_d2sum_kernel(const float* __restrict__ x,
                 const float* __restrict__ y,
                 float* __restrict__ ws) {
    const int z = blockIdx.z;
    const float *A, *B, *sa, *sb;
    select_pair(z, x, y, ws + OFF_SX, ws + OFF_SY, A, B, sa, sb);

    const int wave = blockIdx.x * WAVES_PER_BLOCK + (threadIdx.x >> 5);
    const int r = (wave / NTC) * TILE_R;
    const int c = (wave % NTC) * TILE_C;
    const int lane = threadIdx.x & 31;
    const int m = lane & 15, half = lane >> 4;

    v8f acc[4][2];
    gemm_tile(A, B, r, c, lane, acc);

    float local = 0.f;
#pragma unroll
    for (int qa = 0; qa < 4; ++qa) {
        const int i0 = r + 16 * qa + 8 * half;
        float sav[8];
#pragma unroll
        for (int v = 0; v < 8; ++v) sav[v] = sa[i0 + v];
#pragma unroll
        for (int qb = 0; qb < 2; ++qb) {
            const float sbj = sb[c + 16 * qb + m];
#pragma unroll
            for (int v = 0; v < 8; ++v) {
                float d2 = sav[v] + sbj - 2.f * acc[qa][qb][v];
                local += fmaxf(d2, 0.f);
            }
        }
    }
    block_reduce_store(local, ws + OFF_PART + z * BLOCKS_PER_PAIR + blockIdx.x);
}

// ---- deterministic reduction of block partials per pair ----
__global__ void __launch_bounds__(256)
mmd_reduce_kernel(const float* __restrict__ part,
                  float* __restrict__ out) {
    const int z = blockIdx.x;
    const float* p = part + z * BLOCKS_PER_PAIR;
    float s = 0.f;
    for (int i = threadIdx.x; i < BLOCKS_PER_PAIR; i += 256) s += p[i];
    block_reduce_store(s, out + z);
}

// ---- pass 2: sum over 5 Gaussian kernels ----
__global__ void __launch_bounds__(256)
mmd_kernsum_kernel(const float* __restrict__ x,
                   const float* __restrict__ y,
                   float* __restrict__ ws) {
    const int z = blockIdx.z;
    const float *A, *B, *sa, *sb;
    select_pair(z, x, y, ws + OFF_SX, ws + OFF_SY, A, B, sa, sb);

    // bandwidths from the pass-1 total (stop_gradient path): every thread
    // derives them identically -> deterministic.
    const float nsamp = 2.f * (float)NROWS;                 // 8192
    float bw = ws[z] / (nsamp * nsamp - nsamp);
    bw *= 0.25f;                                            // / 2^(5//2)
    float ib[5];
    {
        float b = bw;
#pragma unroll
        for (int k = 0; k < 5; ++k) { ib[k] = 1.f / (b + EPSK); b *= 2.f; }
    }

    const int wave = blockIdx.x * WAVES_PER_BLOCK + (threadIdx.x >> 5);
    const int r = (wave / NTC) * TILE_R;
    const int c = (wave % NTC) * TILE_C;
    const int lane = threadIdx.x & 31;
    const int m = lane & 15, half = lane >> 4;

    v8f acc[4][2];
    gemm_tile(A, B, r, c, lane, acc);

    float local = 0.f;
#pragma unroll
    for (int qa = 0; qa < 4; ++qa) {
        const int i0 = r + 16 * qa + 8 * half;
        float sav[8];
#pragma unroll
        for (int v = 0; v < 8; ++v) sav[v] = sa[i0 + v];
#pragma unroll
        for (int qb = 0; qb < 2; ++qb) {
            const float sbj = sb[c + 16 * qb + m];
#pragma unroll
            for (int v = 0; v < 8; ++v) {
                float d2 = fmaxf(sav[v] + sbj - 2.f * acc[qa][qb][v], 0.f);
#pragma unroll
                for (int k = 0; k < 5; ++k)
                    local += __expf(-d2 * ib[k]);           // v_exp_f32
            }
        }
    }
    block_reduce_store(local, ws + OFF_PART + z * BLOCKS_PER_PAIR + blockIdx.x);
}

// ---- combine ----
__global__ void __launch_bounds__(32)
mmd_final_kernel(const float* __restrict__ ksum,
                 float* __restrict__ out) {
    if (threadIdx.x == 0) {
        const float n2 = (float)NROWS * (float)NROWS;
        out[0] = (ksum[0] + ksum[1] - 2.f * ksum[2]) / n2;
    }
}

extern "C" void kernel_launch(void* const* d_in, const int* in_sizes, int n_in,
                              void* d_out, int out_size, void* d_ws, size_t ws_size,
                              hipStream_t stream) {
    const float* x = (const float*)d_in[0];
    const float* y = (const float*)d_in[1];
    float* out = (float*)d_out;
    float* ws  = (float*)d_ws;

    // row squared norms: 4096 rows per input, 8 rows/block
    mmd_rownorm_kernel<<<dim3(NROWS / WAVES_PER_BLOCK, 2), 256, 0, stream>>>(x, y, ws);

    dim3 grid(BLOCKS_PER_PAIR, 1, 3);
    // pass 1: d2 sums -> block partials -> totals ws[0..2]
    mmd_d2sum_kernel<<<grid, 256, 0, stream>>>(x, y, ws);
    mmd_reduce_kernel<<<3, 256, 0, stream>>>(ws + OFF_PART, ws);
    // pass 2: kernel sums -> block partials -> totals ws[3..5]
    mmd_kernsum_kernel<<<grid, 256, 0, stream>>>(x, y, ws);
    mmd_reduce_kernel<<<3, 256, 0, stream>>>(ws + OFF_PART, ws + 3);
    mmd_final_kernel<<<1, 32, 0, stream>>>(ws + 3, out);
}